// VQ_28054726377848
// MI455X (gfx1250) — compile-verified
//
#include <hip/hip_runtime.h>
#include <math.h>

typedef float v2f __attribute__((ext_vector_type(2)));
typedef float v4f __attribute__((ext_vector_type(4)));
typedef float v8f __attribute__((ext_vector_type(8)));

#define NTOK 65536   // 16*4096 tokens
#define DIMS 64
#define KCB  512     // codebook size

// ---------------- kernel 1: c_sq precompute + histogram zero ----------------
__global__ void vq_prep(const float* __restrict__ cb,
                        float* __restrict__ csq,
                        int* __restrict__ counts) {
    int i = blockIdx.x * blockDim.x + threadIdx.x;
    if (i < KCB) {
        float s = 0.0f;
        #pragma unroll
        for (int k = 0; k < DIMS; ++k) {
            float c = cb[i * DIMS + k];
            s += c * c;
        }
        csq[i]    = s;
        counts[i] = 0;
    }
}

// ---------------- kernel 2: WMMA distance GEMM + argmin + gather ------------
__global__ __launch_bounds__(256) void vq_main(const float* __restrict__ z,
                                               const float* __restrict__ cb,
                                               const float* __restrict__ csq,
                                               int* __restrict__ counts,
                                               float* __restrict__ zq,
                                               int* __restrict__ qzx,
                                               float* __restrict__ ek) {
    const int lane  = threadIdx.x & 31;
    const int wave  = threadIdx.x >> 5;
    const int w     = blockIdx.x * 8 + wave;   // global wave id, 4096 total
    const int base  = w * 16;                  // first token of this wave's tile
    const int nlane = lane & 15;
    const int half  = lane >> 4;

    // Preload A: 16 tokens x 64 dims, ISA 16x4 f32 A layout per k-step.
    // lane L: a[kk] = { z[base+L%16][4kk + 2*(L/16)], z[...][4kk + 2*(L/16)+1] }
    v2f a[16];
    const float* zrow = z + (size_t)(base + nlane) * DIMS + 2 * half;
    #pragma unroll
    for (int kk = 0; kk < 16; ++kk)
        a[kk] = *(const v2f*)(zrow + 4 * kk);

    float runmin[8];
    int   runidx[8];
    #pragma unroll
    for (int r = 0; r < 8; ++r) { runmin[r] = 3.4e38f; runidx[r] = 0; }

    // 32 column tiles cover all 512 codes
    for (int t = 0; t < 32; ++t) {
        // B (4x16 KxN) lane L: cb[(16t + L%16)][4kk + 2*(L/16) + {0,1}]
        const float* brow = cb + (size_t)(t * 16 + nlane) * DIMS + 2 * half;
        v8f acc = {};
        #pragma unroll
        for (int kk = 0; kk < 16; ++kk) {
            v2f b = *(const v2f*)(brow + 4 * kk);
            // D = A*B + C   (8 args: neg_a, A, neg_b, B, c_mod, C, reuse_a, reuse_b)
            acc = __builtin_amdgcn_wmma_f32_16x16x4_f32(
                false, a[kk], false, b, (short)0, acc, false, false);
        }
        const int   col = t * 16 + nlane;
        const float cn  = csq[col];
        #pragma unroll
        for (int r = 0; r < 8; ++r) {
            float d = cn - 2.0f * acc[r];   // ||c||^2 - 2 z.c  (||z||^2 dropped)
            if (d < runmin[r]) { runmin[r] = d; runidx[r] = col; }
        }
    }

    // Cross-lane argmin within each 16-lane half (wave32; masks stay in-half).
    #pragma unroll
    for (int m = 1; m <= 8; m <<= 1) {
        #pragma unroll
        for (int r = 0; r < 8; ++r) {
            float om = __shfl_xor(runmin[r], m, 32);
            int   oi = __shfl_xor(runidx[r], m, 32);
            if (om < runmin[r] || (om == runmin[r] && oi < runidx[r])) {
                runmin[r] = om; runidx[r] = oi;
            }
        }
    }

    // Write outputs: row M = r + 8*half -> token = base + r + 8*half.
    // All 16 lanes of a half hold identical results; copy 64 floats with 16 lanes.
    #pragma unroll
    for (int r = 0; r < 8; ++r) {
        const int token = base + r + 8 * half;
        const int idx   = runidx[r];
        if (nlane == 0) {
            qzx[token] = idx;
            atomicAdd(&counts[idx], 1);
        }
        v4f cv = *(const v4f*)(cb + (size_t)idx * DIMS + nlane * 4);
        v4f zv = *(const v4f*)(z + (size_t)token * DIMS + nlane * 4);
        v4f zqv;
        #pragma unroll
        for (int j = 0; j < 4; ++j)
            zqv[j] = zv[j] + (cv[j] - zv[j]);   // exact STE rounding: z + (e_k - z)
        *(v4f*)(ek + (size_t)token * DIMS + nlane * 4) = cv;
        *(v4f*)(zq + (size_t)token * DIMS + nlane * 4) = zqv;
    }
}

// ---------------- kernel 3: perplexity ----------------
__global__ void vq_perplex(const int* __restrict__ counts,
                           float* __restrict__ out) {
    __shared__ float s[256];
    const int tid = threadIdx.x;
    float acc = 0.0f;
    for (int i = tid; i < KCB; i += 256) {
        float p = (float)counts[i] * (1.0f / (float)NTOK);
        acc += p * logf(p + 1e-10f);
    }
    s[tid] = acc;
    __syncthreads();
    for (int off = 128; off > 0; off >>= 1) {
        if (tid < off) s[tid] += s[tid + off];
        __syncthreads();
    }
    if (tid == 0) *out = expf(-s[0]);
}

extern "C" void kernel_launch(void* const* d_in, const int* in_sizes, int n_in,
                              void* d_out, int out_size, void* d_ws, size_t ws_size,
                              hipStream_t stream) {
    const float* z  = (const float*)d_in[0];   // (16,4096,64) f32
    const float* cb = (const float*)d_in[1];   // (512,64) f32

    float* out  = (float*)d_out;
    float* zq   = out;                                   // 4194304 f32
    int*   qzx  = (int*)(out + (size_t)NTOK * DIMS);     // 65536 i32
    float* ek   = out + (size_t)NTOK * DIMS + NTOK;      // 4194304 f32
    float* perp = out + 2 * (size_t)NTOK * DIMS + NTOK;  // 1 f32

    float* csq    = (float*)d_ws;                 // 512 f32
    int*   counts = (int*)((char*)d_ws + 2048);   // 512 i32

    vq_prep<<<2, 256, 0, stream>>>(cb, csq, counts);
    vq_main<<<512, 256, 0, stream>>>(z, cb, csq, counts, zq, qzx, ek);
    vq_perplex<<<1, 256, 0, stream>>>(counts, perp);
}